// LogEig_42898133352585
// MI455X (gfx1250) — compile-verified
//
#include <hip/hip_runtime.h>
#include <math.h>

// LogEig for 8192 batched 64x64 SPD matrices via inverse scaling-and-squaring:
//   A1 = P / ||P||_F                        (eigenvalues in (0,1])
//   repeat k=6: A1 <- sqrt(A1)              (coupled Newton-Schulz, matmul-only)
//   log(P) = log(c) I + 2^k * log(A1),  log(I+M) via Horner Mercator series.
// All matmuls on v_wmma_f32_16x16x4_f32; matrices resident in LDS.
// LDS row stride 76 floats: A(b64)/B/C accesses all provably bank-conflict-free.

typedef float v2f __attribute__((ext_vector_type(2)));
typedef float v8f __attribute__((ext_vector_type(8)));

#define LSTR      76              // LDS row stride (floats)
#define BUF       (64 * LSTR)     // floats per 64x64 LDS matrix buffer
#define SH_FLOATS (4 * BUF + 256) // 4 matrices + reduction scratch
#define SH_BYTES  (SH_FLOATS * sizeof(float))

#define SQRT_STEPS 6
#define LOG_TERMS  12

// Core: one wave computes two adjacent 16x16 tiles (same tile-row, shared A frag).
// acc0/acc1 = A_tile_row @ B_tile_cols; returns via reference.
__device__ __forceinline__ void wave_tiles(const float* A, const float* B,
                                           int ti, int tj0, int lane,
                                           v8f& acc0, v8f& acc1)
{
  const int m  = lane & 15;           // M for A-frag, N for B-frag
  const int kk = (lane >> 4) << 1;    // K base per half-wave (0 or 2)
  const float* pa = A + (ti * 16 + m) * LSTR + kk;
  const float* pb = B + kk * LSTR + m;
  const int c0 = tj0 * 16, c1 = c0 + 16;
#pragma unroll
  for (int kb = 0; kb < 64; kb += 4) {
    v2f a = *(const v2f*)(pa + kb);                       // ds_load_b64, shared
    v2f b0, b1;
    b0.x = pb[kb * LSTR + c0]; b0.y = pb[(kb + 1) * LSTR + c0];
    b1.x = pb[kb * LSTR + c1]; b1.y = pb[(kb + 1) * LSTR + c1];
    acc0 = __builtin_amdgcn_wmma_f32_16x16x4_f32(false, a, false, b0,
                                                 (short)0, acc0, false, false);
    acc1 = __builtin_amdgcn_wmma_f32_16x16x4_f32(false, a, false, b1,
                                                 (short)0, acc1, false, false);
  }
}

// D = alpha * (A @ B) + diag * I   (64x64 in LDS, stride LSTR)
__device__ __forceinline__ void mm64(const float* A, const float* B, float* D,
                                     float alpha, float diag, int lane, int wave)
{
  const int t0 = wave * 2;            // 8 waves x 2 tiles = 16 tiles; same ti pair
  const int ti = t0 >> 2, tj0 = t0 & 3;
  v8f acc0 = {0.f,0.f,0.f,0.f,0.f,0.f,0.f,0.f};
  v8f acc1 = {0.f,0.f,0.f,0.f,0.f,0.f,0.f,0.f};
  wave_tiles(A, B, ti, tj0, lane, acc0, acc1);

  const int m   = lane & 15;
  const int r0  = ti * 16 + ((lane >> 4) << 3);
  const int c0  = tj0 * 16 + m, c1 = c0 + 16;
#pragma unroll
  for (int r = 0; r < 8; ++r) {
    const int row = r0 + r;
    D[row * LSTR + c0] = alpha * acc0[r] + ((row == c0) ? diag : 0.f);
    D[row * LSTR + c1] = alpha * acc1[r] + ((row == c1) ? diag : 0.f);
  }
}

// G(64x64 row-major, global) = alpha * (A @ B) + diag * I
__device__ __forceinline__ void mm64_out(const float* A, const float* B,
                                         float* __restrict__ G,
                                         float alpha, float diag, int lane, int wave)
{
  const int t0 = wave * 2;
  const int ti = t0 >> 2, tj0 = t0 & 3;
  v8f acc0 = {0.f,0.f,0.f,0.f,0.f,0.f,0.f,0.f};
  v8f acc1 = {0.f,0.f,0.f,0.f,0.f,0.f,0.f,0.f};
  wave_tiles(A, B, ti, tj0, lane, acc0, acc1);

  const int m   = lane & 15;
  const int r0  = ti * 16 + ((lane >> 4) << 3);
  const int c0  = tj0 * 16 + m, c1 = c0 + 16;
#pragma unroll
  for (int r = 0; r < 8; ++r) {
    const int row = r0 + r;
    G[row * 64 + c0] = alpha * acc0[r] + ((row == c0) ? diag : 0.f);
    G[row * 64 + c1] = alpha * acc1[r] + ((row == c1) ? diag : 0.f);
  }
}

__global__ __launch_bounds__(256) void logeig_kernel(const float* __restrict__ P,
                                                     float* __restrict__ out)
{
  extern __shared__ float smem[];
  float* b0  = smem;
  float* b1  = smem + BUF;
  float* b2  = smem + 2 * BUF;
  float* b3  = smem + 3 * BUF;
  float* red = smem + 4 * BUF;

  const int tid  = threadIdx.x;
  const int lane = tid & 31;
  const int wave = tid >> 5;
  const size_t base = (size_t)blockIdx.x * 4096;

  // Each thread owns a quarter-row: row = tid>>2, cols [(tid&3)*16, +16)
  const int myrow = tid >> 2;
  const int mycol = (tid & 3) * 16;

  // ---- load P -> Y(b0) (coalesced b128), fused Frobenius norm^2 ----
  {
    const float4* src = (const float4*)(P + base);
    float s = 0.f;
#pragma unroll
    for (int i = 0; i < 4; ++i) {
      float4 v = src[tid * 4 + i];
      float* dst = &b0[myrow * LSTR + mycol + i * 4];
      dst[0] = v.x; dst[1] = v.y; dst[2] = v.z; dst[3] = v.w;
      s += v.x * v.x + v.y * v.y + v.z * v.z + v.w * v.w;
    }
    red[tid] = s;
  }
  __syncthreads();
  for (int off = 128; off > 0; off >>= 1) {
    if (tid < off) red[tid] += red[tid + off];
    __syncthreads();
  }
  const float fro  = sqrtf(red[0]);
  const float rinv = 1.0f / fro;

  // scale Y by 1/c
#pragma unroll
  for (int i = 0; i < 16; ++i)
    b0[myrow * LSTR + mycol + i] *= rinv;
  __syncthreads();

  // ---- k chained Newton-Schulz square roots ----
  float* Y = b0; float* Z = b1; float* T = b2; float* U = b3;
  const int iters[SQRT_STEPS] = {16, 13, 11, 9, 8, 7};

#pragma unroll 1
  for (int sq = 0; sq < SQRT_STEPS; ++sq) {
    // Z = I for this sqrt stage
#pragma unroll
    for (int i = 0; i < 16; ++i)
      Z[myrow * LSTR + mycol + i] = (myrow == mycol + i) ? 1.f : 0.f;
    __syncthreads();

#pragma unroll 1
    for (int it = 0; it < iters[sq]; ++it) {
      mm64(Z, Y, T, -0.5f, 1.5f, lane, wave);   // T = 1.5I - 0.5*Z@Y
      __syncthreads();
      mm64(Y, T, U, 1.0f, 0.0f, lane, wave);    // Ynew = Y@T
      __syncthreads();                          // all waves done reading Y
      mm64(T, Z, Y, 1.0f, 0.0f, lane, wave);    // Znew = T@Z  (into old Y)
      __syncthreads();
      float* nY = U; float* nZ = Y;             // rotate buffers
      U = Z; Y = nY; Z = nZ;
    }
  }

  // ---- log(I+M) by Horner Mercator series, M = Y - I (into T) ----
#pragma unroll
  for (int i = 0; i < 16; ++i) {
    const int c = mycol + i;
    T[myrow * LSTR + c] = Y[myrow * LSTR + c] - ((myrow == c) ? 1.f : 0.f);
  }
  // R = c_q * I
  {
    const float cq = ((LOG_TERMS & 1) ? 1.0f : -1.0f) / (float)LOG_TERMS;
#pragma unroll
    for (int i = 0; i < 16; ++i)
      Z[myrow * LSTR + mycol + i] = (myrow == mycol + i) ? cq : 0.f;
  }
  __syncthreads();

  float* R = Z; float* Rn = U;
#pragma unroll 1
  for (int j = LOG_TERMS - 1; j >= 1; --j) {
    const float cj = ((j & 1) ? 1.0f : -1.0f) / (float)j;
    mm64(T, R, Rn, 1.0f, cj, lane, wave);       // R = cj*I + M@R
    __syncthreads();
    float* tmp = R; R = Rn; Rn = tmp;
  }

  // out = 2^k * (M@R) + log(c) * I
  mm64_out(T, R, out + base, (float)(1 << SQRT_STEPS), logf(fro), lane, wave);
}

extern "C" void kernel_launch(void* const* d_in, const int* in_sizes, int n_in,
                              void* d_out, int out_size, void* d_ws, size_t ws_size,
                              hipStream_t stream) {
  (void)n_in; (void)d_ws; (void)ws_size; (void)out_size;
  const float* P = (const float*)d_in[0];
  float* out = (float*)d_out;
  const int nmat = in_sizes[0] / 4096;   // B*H = 8192 matrices of 64x64

  // ~78 KB dynamic LDS per block (WGP has 320 KB -> 4 blocks resident)
  static_assert(SH_BYTES < 320 * 1024, "LDS over WGP limit");
  hipFuncSetAttribute((const void*)logeig_kernel,
                      hipFuncAttributeMaxDynamicSharedMemorySize, (int)SH_BYTES);
  logeig_kernel<<<dim3(nmat), dim3(256), SH_BYTES, stream>>>(P, out);
}